// GRUTopK_28767690949408
// MI455X (gfx1250) — compile-verified
//
#include <hip/hip_runtime.h>
#include <math.h>

#define R_NODES 100000
#define DIM 512
#define NELEM (DIM * DIM)

typedef float v2f __attribute__((ext_vector_type(2)));
typedef float v8f __attribute__((ext_vector_type(8)));

// ---------------- workspace layout (bytes) ----------------
#define WS_KEYS   0u            // R_NODES u64         (800000 B)
#define WS_NORM   819200u       // 1 float
#define WS_STATE  819216u       // RState (16 B)
#define WS_HIST   819264u       // 256 u32             (1024 B)
#define WS_CAND   820288u       // 512 u64             (4096 B)
#define WS_IDX    824384u       // 512 int
#define WS_TANH   826432u       // 512 float
#define WS_XT     (1u << 20)    // 512x512 f32 (x_topk^T)
#define WS_HT     (2u << 20)    // 512x512 f32 (hist^T)
#define WS_PREU   (3u << 20)    // pre-activation update -> update
#define WS_PRER   (4u << 20)    // pre-activation reset
#define WS_RHT    (5u << 20)    // (reset*hist)^T
#define WS_PREH   (6u << 20)    // pre-activation h_cap

struct RState {
  unsigned long long prefix;
  unsigned int remaining;
  unsigned int cnt;
};

__device__ __forceinline__ unsigned sortable_u32(float f) {
  unsigned u = __float_as_uint(f);
  return (u & 0x80000000u) ? ~u : (u | 0x80000000u);
}
__device__ __forceinline__ float unsortable_f32(unsigned s) {
  unsigned u = (s & 0x80000000u) ? (s & 0x7FFFFFFFu) : ~s;
  return __uint_as_float(u);
}
__device__ __forceinline__ float sigmoid_f(float x) { return 1.0f / (1.0f + expf(-x)); }

// ---------------- ||scorer|| ----------------
__global__ void k_norm(const float* __restrict__ scorer, float* __restrict__ normOut) {
  __shared__ float red[256];
  int t = threadIdx.x;
  float a = scorer[t];
  float b = scorer[t + 256];
  red[t] = a * a + b * b;
  __syncthreads();
  for (int s = 128; s > 0; s >>= 1) {
    if (t < s) red[t] += red[t + s];
    __syncthreads();
  }
  if (t == 0) normOut[0] = sqrtf(red[0]) + 1e-12f;
}

// ---------------- scores: one wave32 per row, coalesced float4 ----------------
__global__ void k_scores(const float* __restrict__ inputs, const float* __restrict__ mask,
                         const float* __restrict__ scorer, const float* __restrict__ norm,
                         unsigned long long* __restrict__ keys) {
  int lane = threadIdx.x & 31;
  int row = blockIdx.x * 8 + (threadIdx.x >> 5);
  if (row >= R_NODES) return;
  const float4* p = (const float4*)(inputs + (size_t)row * DIM);
  const float4* q = (const float4*)scorer;
  float s = 0.0f;
#pragma unroll
  for (int j = 0; j < 4; ++j) {
    float4 v = p[lane + j * 32];
    float4 w = q[lane + j * 32];
    s += v.x * w.x + v.y * w.y + v.z * w.z + v.w * w.w;
  }
#pragma unroll
  for (int off = 16; off > 0; off >>= 1) s += __shfl_xor(s, off, 32);
  if (lane == 0) {
    float score = s / norm[0] + mask[row];
    unsigned k32 = sortable_u32(score);
    // low 32 bits: ~row -> all keys distinct, ties resolved to lower index first
    keys[row] = ((unsigned long long)k32 << 32) | (unsigned)(~row);
  }
}

// ---------------- radix select (8 x 8-bit digits, MSB first) ----------------
__global__ void k_radix_init(unsigned* __restrict__ hist, RState* __restrict__ st) {
  hist[threadIdx.x] = 0u;
  if (threadIdx.x == 0) {
    st->prefix = 0ull;
    st->remaining = 512u;
    st->cnt = 0u;
  }
}

__global__ void k_radix_hist(const unsigned long long* __restrict__ keys,
                             unsigned* __restrict__ hist, const RState* __restrict__ st,
                             int pass) {
  unsigned long long prefix = st->prefix;
  unsigned long long maskHigh = (pass == 0) ? 0ull : (~0ull << (64 - 8 * pass));
  int shift = 56 - 8 * pass;
  for (int i = blockIdx.x * blockDim.x + threadIdx.x; i < R_NODES;
       i += gridDim.x * blockDim.x) {
    unsigned long long k = keys[i];
    if ((k & maskHigh) == prefix) {
      unsigned d = (unsigned)((k >> shift) & 0xFFull);
      atomicAdd(&hist[d], 1u);
    }
  }
}

__global__ void k_radix_scan(unsigned* __restrict__ hist, RState* __restrict__ st, int pass) {
  if (threadIdx.x == 0) {
    int shift = 56 - 8 * pass;
    unsigned rem = st->remaining;
    unsigned cum = 0;
    unsigned long long pre = st->prefix;
    for (int b = 255; b >= 0; --b) {
      unsigned c = hist[b];
      if (cum + c >= rem) {
        pre |= ((unsigned long long)(unsigned)b) << shift;
        rem -= cum;
        break;
      }
      cum += c;
    }
    st->prefix = pre;
    st->remaining = rem;
  }
  __syncthreads();
  hist[threadIdx.x] = 0u;  // ready for next pass
}

__global__ void k_compact(const unsigned long long* __restrict__ keys, RState* __restrict__ st,
                          unsigned long long* __restrict__ cand) {
  unsigned long long thr = st->prefix;  // exact 512th-largest key (keys distinct)
  for (int i = blockIdx.x * blockDim.x + threadIdx.x; i < R_NODES;
       i += gridDim.x * blockDim.x) {
    unsigned long long k = keys[i];
    if (k >= thr) {
      unsigned p = atomicAdd(&st->cnt, 1u);
      if (p < 512u) cand[p] = k;
    }
  }
}

// ---------------- bitonic sort 512 keys (descending) + extract idx/tanh ----------------
__global__ void k_sort_extract(const unsigned long long* __restrict__ cand,
                               int* __restrict__ idxOut, float* __restrict__ tanhOut) {
  __shared__ unsigned long long s[512];
  unsigned tid = threadIdx.x;
  s[tid] = cand[tid];
  __syncthreads();
  for (unsigned k = 2; k <= 512; k <<= 1) {
    for (unsigned j = k >> 1; j > 0; j >>= 1) {
      unsigned ixj = tid ^ j;
      if (ixj > tid) {
        unsigned long long a = s[tid];
        unsigned long long b = s[ixj];
        bool descending = ((tid & k) == 0);
        if (descending ? (a < b) : (a > b)) {
          s[tid] = b;
          s[ixj] = a;
        }
      }
      __syncthreads();
    }
  }
  unsigned long long key = s[tid];
  unsigned row = ~(unsigned)key;
  float val = unsortable_f32((unsigned)(key >> 32));
  idxOut[tid] = (int)row;
  tanhOut[tid] = tanhf(val);
}

// ---------------- xt[n][k] = inputs[idx[n]][k] * tanh(vals[n])  (= x_topk^T) ----------------
__global__ void k_build_xt(const float* __restrict__ inputs, const int* __restrict__ idx,
                           const float* __restrict__ tanhv, float* __restrict__ xt) {
  int i = blockIdx.x * blockDim.x + threadIdx.x;  // 65536 threads, one float4 each
  int n = i >> 7;
  int k4 = (i & 127) << 2;
  float t = tanhv[n];
  float4 v = *(const float4*)(inputs + (size_t)idx[n] * DIM + k4);
  float4 o;
  o.x = v.x * t; o.y = v.y * t; o.z = v.z * t; o.w = v.w * t;
  *(float4*)(xt + n * DIM + k4) = o;
}

// ---------------- dst = src^T (512x512) ----------------
__global__ void k_transpose(const float* __restrict__ src, float* __restrict__ dst) {
  int i = blockIdx.x * blockDim.x + threadIdx.x;  // exactly NELEM threads
  int n = i >> 9;
  int k = i & 511;
  dst[i] = src[(k << 9) + n];
}

// ---------------- OUT = A1 @ B1^T' + A2 @ B2^T' + bias via V_WMMA_F32_16X16X4_F32 ----
// A1/A2: 512x512 row-major. BT1/BT2 hold B transposed: BT[n][k] = B[k][n].
// Wave computes a 16(M) x 64(N) strip = 4 accumulator tiles; block = 8 waves = 128x64.
__global__ void k_gemm(const float* __restrict__ A1, const float* __restrict__ A2,
                       const float* __restrict__ bias, const float* __restrict__ BT1,
                       const float* __restrict__ BT2, float* __restrict__ out) {
  const int lane = threadIdx.x & 31;
  const int wave = threadIdx.x >> 5;
  const int tm = blockIdx.y * 128 + wave * 16;
  const int tnBase = blockIdx.x * 64;
  const int lm = lane & 15;           // row/col within fragment
  const int koff = (lane >> 4) << 1;  // K offset: lanes 16-31 hold K+2,K+3
  const int mOff = (lane >> 4) << 3;  // C/D: lanes 16-31 hold M+8

  v8f acc[4];
#pragma unroll
  for (int t = 0; t < 4; ++t) {
    const int tn = tnBase + t * 16;
#pragma unroll
    for (int v = 0; v < 8; ++v) acc[t][v] = bias[(tm + v + mOff) * DIM + tn + lm];
  }

  const float* A = A1;
  const float* BT = BT1;
#pragma unroll 1
  for (int phase = 0; phase < 2; ++phase) {
    for (int k0 = 0; k0 < DIM; k0 += 4) {
      const int kk = k0 + koff;
      v2f a = *(const v2f*)(A + (tm + lm) * DIM + kk);  // A[m][kk], A[m][kk+1]
#pragma unroll
      for (int t = 0; t < 4; ++t) {
        const int tn = tnBase + t * 16;
        v2f b = *(const v2f*)(BT + (tn + lm) * DIM + kk);  // B[kk][n], B[kk+1][n]
        acc[t] = __builtin_amdgcn_wmma_f32_16x16x4_f32(false, a, false, b, (short)0,
                                                       acc[t], false, false);
      }
    }
    A = A2;
    BT = BT2;
  }

#pragma unroll
  for (int t = 0; t < 4; ++t) {
    const int tn = tnBase + t * 16;
#pragma unroll
    for (int v = 0; v < 8; ++v) out[(tm + v + mOff) * DIM + tn + lm] = acc[t][v];
  }
}

// ---------------- elementwise: update = sigmoid(preU); rht = (sigmoid(preR)*hist)^T ---
__global__ void k_elem1(float* __restrict__ preU, const float* __restrict__ preR,
                        const float* __restrict__ hist, float* __restrict__ rht) {
  int i = blockIdx.x * blockDim.x + threadIdx.x;  // exactly NELEM threads
  int k = i >> 9;
  int n = i & 511;
  float u = sigmoid_f(preU[i]);
  preU[i] = u;  // becomes `update`
  float r = sigmoid_f(preR[i]);
  rht[(n << 9) + k] = r * hist[i];
}

// ---------------- out = (1-update)*hist + update*tanh(preH) ----------------
__global__ void k_elem2(const float* __restrict__ upd, const float* __restrict__ preH,
                        const float* __restrict__ hist, float* __restrict__ out) {
  int i = blockIdx.x * blockDim.x + threadIdx.x;
  float u = upd[i];
  out[i] = (1.0f - u) * hist[i] + u * tanhf(preH[i]);
}

extern "C" void kernel_launch(void* const* d_in, const int* in_sizes, int n_in,
                              void* d_out, int out_size, void* d_ws, size_t ws_size,
                              hipStream_t stream) {
  (void)in_sizes; (void)n_in; (void)out_size; (void)ws_size;
  const float* inputs = (const float*)d_in[0];
  const float* hist   = (const float*)d_in[1];
  const float* mask   = (const float*)d_in[2];
  const float* scorer = (const float*)d_in[3];
  const float* Wu = (const float*)d_in[4];
  const float* Uu = (const float*)d_in[5];
  const float* bu = (const float*)d_in[6];
  const float* Wr = (const float*)d_in[7];
  const float* Ur = (const float*)d_in[8];
  const float* br = (const float*)d_in[9];
  const float* Wh = (const float*)d_in[10];
  const float* Uh = (const float*)d_in[11];
  const float* bh = (const float*)d_in[12];
  float* out = (float*)d_out;
  char* ws = (char*)d_ws;

  unsigned long long* keys = (unsigned long long*)(ws + WS_KEYS);
  float* norm = (float*)(ws + WS_NORM);
  RState* st = (RState*)(ws + WS_STATE);
  unsigned* hbins = (unsigned*)(ws + WS_HIST);
  unsigned long long* cand = (unsigned long long*)(ws + WS_CAND);
  int* idx = (int*)(ws + WS_IDX);
  float* tanhv = (float*)(ws + WS_TANH);
  float* xt   = (float*)(ws + WS_XT);
  float* ht   = (float*)(ws + WS_HT);
  float* preU = (float*)(ws + WS_PREU);
  float* preR = (float*)(ws + WS_PRER);
  float* rht  = (float*)(ws + WS_RHT);
  float* preH = (float*)(ws + WS_PREH);

  k_norm<<<1, 256, 0, stream>>>(scorer, norm);
  k_scores<<<(R_NODES + 7) / 8, 256, 0, stream>>>(inputs, mask, scorer, norm, keys);

  k_radix_init<<<1, 256, 0, stream>>>(hbins, st);
  for (int p = 0; p < 8; ++p) {
    k_radix_hist<<<391, 256, 0, stream>>>(keys, hbins, st, p);
    k_radix_scan<<<1, 256, 0, stream>>>(hbins, st, p);
  }
  k_compact<<<391, 256, 0, stream>>>(keys, st, cand);
  k_sort_extract<<<1, 512, 0, stream>>>(cand, idx, tanhv);

  k_build_xt<<<256, 256, 0, stream>>>(inputs, idx, tanhv, xt);
  k_transpose<<<NELEM / 256, 256, 0, stream>>>(hist, ht);

  dim3 gg(DIM / 64, DIM / 128);
  k_gemm<<<gg, 256, 0, stream>>>(Wu, Uu, bu, xt, ht, preU);
  k_gemm<<<gg, 256, 0, stream>>>(Wr, Ur, br, xt, ht, preR);
  k_elem1<<<NELEM / 256, 256, 0, stream>>>(preU, preR, hist, rht);
  k_gemm<<<gg, 256, 0, stream>>>(Wh, Uh, bh, xt, rht, preH);
  k_elem2<<<NELEM / 256, 256, 0, stream>>>(preU, preH, hist, out);
}